// TopologicalAutoencoder_75909251989525
// MI455X (gfx1250) — compile-verified
//
#include <hip/hip_runtime.h>

// Topological Autoencoder forward on MI455X (gfx1250):
//   Z     = X @ W_enc + b_enc        [16384 x 64]
//   preds = Z @ W_dec + b_dec        [16384 x 1024]
//   loss  = 0.5 * sum_e (||X_a - X_b|| - ||Z_a - Z_b||)^2 over 2E edges
//
// fp32 WMMA (V_WMMA_F32_16X16X4_F32) keeps fp32 semantics; workload is
// HBM-bound (~132 MB / 23.3 TB/s ~= 5.7 us). Weight matrices are staged into
// LDS with the Tensor Data Mover (tensor_load_to_lds + s_wait_tensorcnt),
// descriptor built per CDNA5 ISA. Each wave computes a
// 16x64 output strip so one A fragment feeds 4 WMMAs.

#define NPTS   16384
#define DDIM   1024
#define NLAT   64
#define EEDGES 32768
#define KC     64          // encoder K-chunk staged per TDM transfer

typedef __attribute__((ext_vector_type(2))) float v2f;
typedef __attribute__((ext_vector_type(4))) float v4f;
typedef __attribute__((ext_vector_type(8))) float v8f;
typedef unsigned int u32x4 __attribute__((ext_vector_type(4)));
typedef int          i32x4 __attribute__((ext_vector_type(4)));
typedef int          i32x8 __attribute__((ext_vector_type(8)));

// Generic pointers to LDS carry the LDS byte offset in their low 32 bits.
__device__ __forceinline__ uint32_t lds_off_of(const void* p) {
    return (uint32_t)(uintptr_t)p;
}

// ---------------------------------------------------------------------------
// TDM 2D tile load: tile_y rows of tile_x fp32 elements, row stride
// row_stride_elems, from `g` (tile start) into LDS at lds_byte_off.
// D# packing per ISA §8.3/§8.4:
//   group0: count=1 | lds_addr | global_addr[56:0] | type=2
//   group1: data_size=2(4B); tensor_dim0/1 = tile dims (tile fully in-bounds);
//           tile_dim0/1; tensor_dim0_stride; tile_dim2=0 (2D)
//   group2/3 (and trailing vector): zeros (2D tensor)
// Toolchain here is the 6-arg amdgpu-toolchain form:
//   (u32x4 g0, i32x8 g1, i32x4 g2, i32x4 g3, i32x8, i32 cpol)
// ---------------------------------------------------------------------------
__device__ __forceinline__ void tdm_load_2d(uint32_t lds_byte_off,
                                            const float* g,
                                            uint32_t tile_x, uint32_t tile_y,
                                            uint32_t row_stride_elems) {
    const uint64_t ga = (uint64_t)(uintptr_t)g;
    u32x4 g0;
    g0.x = 1u;                                    // count=1, user mode, no gather
    g0.y = lds_byte_off;                          // lds_addr
    g0.z = (uint32_t)ga;                          // global_addr[31:0]
    g0.w = (uint32_t)((ga >> 32) & 0x1FFFFFFu)    // global_addr[56:32]
         | (2u << 30);                            // type=2 ("image")
    i32x8 g1;
    g1[0] = (int)(2u << 16);                      // data_size=2 -> 4 bytes
    g1[1] = (int)(tile_x << 16);                  // tensor_dim0[15:0] @ [63:48]
    g1[2] = (int)((tile_x >> 16) | (tile_y << 16)); // tensor_dim0 hi | tensor_dim1 lo
    g1[3] = (int)((tile_y >> 16) | (tile_x << 16)); // tensor_dim1 hi | tile_dim0
    g1[4] = (int)tile_y;                          // tile_dim1 ; tile_dim2 = 0
    g1[5] = (int)row_stride_elems;                // tensor_dim0_stride[31:0]
    g1[6] = 0;                                    // stride0 hi | stride1 lo
    g1[7] = 0;                                    // stride1 hi
    const i32x4 z4 = {0, 0, 0, 0};
    const i32x8 z8 = {0, 0, 0, 0, 0, 0, 0, 0};
    __builtin_amdgcn_tensor_load_to_lds(g0, g1, z4, z4, z8, 0);
}

// ---------------------------------------------------------------------------
// Encoder: Z[16384 x 64] = X @ W_enc + b_enc.
// Block: 8 waves x 16 rows = 128 rows; each wave computes 16 rows x 64 cols
// (4 accumulators). W_enc streamed through LDS as 16 double-buffered 64x64
// chunks via TDM (wave 0 issues, TENSORcnt + barrier syncs).
// Per-lane WMMA layouts (ISA 7.12.2, f32, wave32):
//   A 16x4: lane loads float2 at A[m] + k + 2*(lane>>4)
//   B 4x16: VGPR0 = rows {k, k+2}, VGPR1 = rows {k+1, k+3}
//   C:      VGPR v = rows {v, v+8} split across lane halves
// ---------------------------------------------------------------------------
__global__ __launch_bounds__(256)
void encoder_kernel(const float* __restrict__ X,
                    const float* __restrict__ W,
                    const float* __restrict__ bias,
                    float* __restrict__ Z) {
    __shared__ float bbuf[2][KC * NLAT];          // 2 x 16 KB
    const int lane = threadIdx.x & 31;
    const int wave = threadIdx.x >> 5;
    const int half = lane >> 4;
    const int l15  = lane & 15;
    const int mRow = blockIdx.x * 128 + wave * 16 + l15;
    const float* __restrict__ aRow = X + (size_t)mRow * DDIM;

    if (wave == 0) {
        tdm_load_2d(lds_off_of(&bbuf[0][0]), W, NLAT, KC, NLAT);
        __builtin_amdgcn_s_wait_tensorcnt(0);
    }
    __syncthreads();

    float bval[4];
    v8f   acc[4];
#pragma unroll
    for (int t = 0; t < 4; ++t) {
        bval[t] = bias[t * 16 + l15];
        acc[t]  = (v8f){};
    }

    constexpr int nChunks = DDIM / KC;            // 16
    for (int j = 0; j < nChunks; ++j) {
        const int cur = j & 1;
        if (wave == 0 && (j + 1) < nChunks) {     // prefetch next chunk (no wait yet)
            tdm_load_2d(lds_off_of(&bbuf[cur ^ 1][0]),
                        W + (size_t)(j + 1) * KC * NLAT, NLAT, KC, NLAT);
        }
        const float* __restrict__ ldsB = &bbuf[cur][0];
#pragma unroll 4
        for (int kk = 0; kk < KC; kk += 4) {
            v2f a = *(const v2f*)(aRow + j * KC + kk + 2 * half);
#pragma unroll
            for (int t = 0; t < 4; ++t) {
                v2f b;
                b.x = ldsB[(kk + 2 * half)     * NLAT + t * 16 + l15];
                b.y = ldsB[(kk + 2 * half + 1) * NLAT + t * 16 + l15];
                acc[t] = __builtin_amdgcn_wmma_f32_16x16x4_f32(
                             false, a, false, b, (short)0, acc[t], false, false);
            }
        }
        if (wave == 0) __builtin_amdgcn_s_wait_tensorcnt(0);
        __syncthreads();
    }

#pragma unroll
    for (int t = 0; t < 4; ++t) {
        float* __restrict__ cBase =
            Z + (size_t)(blockIdx.x * 128 + wave * 16 + half * 8) * NLAT + t * 16 + l15;
#pragma unroll
        for (int v = 0; v < 8; ++v)
            cBase[(size_t)v * NLAT] = acc[t][v] + bval[t];
    }
}

// ---------------------------------------------------------------------------
// Decoder: preds[16384 x 1024] = Z @ W_dec + b_dec.
// Block: 128 rows x 64 cols. The 64x64 W_dec slice (16 KB, row stride 1024)
// is a single TDM 2D-tile load; K-loop is fully LDS-fed.
// ---------------------------------------------------------------------------
__global__ __launch_bounds__(256)
void decoder_kernel(const float* __restrict__ Zin,
                    const float* __restrict__ W,
                    const float* __restrict__ bias,
                    float* __restrict__ P) {
    __shared__ float bbuf[NLAT * 64];             // 16 KB: 64 K-rows x 64 cols
    const int lane   = threadIdx.x & 31;
    const int wave   = threadIdx.x >> 5;
    const int half   = lane >> 4;
    const int l15    = lane & 15;
    const int colGrp = blockIdx.x & 15;           // 16 column groups of 64
    const int rowBlk = blockIdx.x >> 4;           // 128 row blocks of 128
    const int c0     = colGrp * 64;
    const int mRow   = rowBlk * 128 + wave * 16 + l15;

    if (wave == 0) {
        tdm_load_2d(lds_off_of(&bbuf[0]), W + c0, 64, NLAT, DDIM);
        __builtin_amdgcn_s_wait_tensorcnt(0);
    }
    __syncthreads();

    const float* __restrict__ aRow = Zin + (size_t)mRow * NLAT;
    float bval[4];
    v8f   acc[4];
#pragma unroll
    for (int t = 0; t < 4; ++t) {
        bval[t] = bias[c0 + t * 16 + l15];
        acc[t]  = (v8f){};
    }

#pragma unroll 4
    for (int kk = 0; kk < NLAT; kk += 4) {
        v2f a = *(const v2f*)(aRow + kk + 2 * half);
#pragma unroll
        for (int t = 0; t < 4; ++t) {
            v2f b;
            b.x = bbuf[(kk + 2 * half)     * 64 + t * 16 + l15];
            b.y = bbuf[(kk + 2 * half + 1) * 64 + t * 16 + l15];
            acc[t] = __builtin_amdgcn_wmma_f32_16x16x4_f32(
                         false, a, false, b, (short)0, acc[t], false, false);
        }
    }

#pragma unroll
    for (int t = 0; t < 4; ++t) {
        float* __restrict__ cBase =
            P + (size_t)(rowBlk * 128 + wave * 16 + half * 8) * DDIM + c0 + t * 16 + l15;
#pragma unroll
        for (int v = 0; v < 8; ++v)
            cBase[(size_t)v * DDIM] = acc[t][v] + bval[t];
    }
}

// ---------------------------------------------------------------------------
// Topological loss: one wave per edge, coalesced float4 gathers over D=1024
// (X and Z are L2-resident: 64 MB + 4 MB << 192 MB L2), wave32 butterfly
// reduction, per-block LDS combine, one atomicAdd per block.
// ---------------------------------------------------------------------------
__global__ __launch_bounds__(256)
void topo_loss_kernel(const float* __restrict__ X,
                      const float* __restrict__ Z,
                      const int* __restrict__ pi_x,
                      const int* __restrict__ pi_z,
                      float* __restrict__ loss_out) {
    const int lane = threadIdx.x & 31;
    const int wave = threadIdx.x >> 5;
    const int edge = blockIdx.x * (blockDim.x >> 5) + wave;

    __shared__ float partial[8];

    float contrib = 0.0f;
    if (edge < 2 * EEDGES) {
        const int* ep = (edge < EEDGES) ? (pi_x + 2 * (size_t)edge)
                                        : (pi_z + 2 * (size_t)(edge - EEDGES));
        const int n1 = ep[0];
        const int n2 = ep[1];

        const v4f* __restrict__ x1 = (const v4f*)(X + (size_t)n1 * DDIM);
        const v4f* __restrict__ x2 = (const v4f*)(X + (size_t)n2 * DDIM);
        float sx = 0.0f;
#pragma unroll
        for (int i = 0; i < DDIM / (4 * 32); ++i) {   // 8 iterations
            v4f d = x1[lane + i * 32] - x2[lane + i * 32];
            sx += d.x * d.x + d.y * d.y + d.z * d.z + d.w * d.w;
        }

        const v2f* __restrict__ z1 = (const v2f*)(Z + (size_t)n1 * NLAT);
        const v2f* __restrict__ z2 = (const v2f*)(Z + (size_t)n2 * NLAT);
        v2f dz = z1[lane] - z2[lane];
        float sz = dz.x * dz.x + dz.y * dz.y;

#pragma unroll
        for (int off = 16; off > 0; off >>= 1) {
            sx += __shfl_xor(sx, off, 32);
            sz += __shfl_xor(sz, off, 32);
        }

        const float d = sqrtf(sx) - sqrtf(sz);
        contrib = 0.5f * d * d;
    }

    if (lane == 0) partial[wave] = contrib;
    __syncthreads();

    if (threadIdx.x == 0) {
        float s = 0.0f;
        const int nw = blockDim.x >> 5;
        for (int w = 0; w < nw; ++w) s += partial[w];
        atomicAdd(loss_out, s);
    }
}

__global__ void zero_loss_kernel(float* __restrict__ loss) { *loss = 0.0f; }

// ---------------------------------------------------------------------------
extern "C" void kernel_launch(void* const* d_in, const int* in_sizes, int n_in,
                              void* d_out, int out_size, void* d_ws, size_t ws_size,
                              hipStream_t stream) {
    (void)in_sizes; (void)n_in; (void)out_size; (void)ws_size;

    const float* X     = (const float*)d_in[0];
    const float* W_enc = (const float*)d_in[1];
    const float* b_enc = (const float*)d_in[2];
    const float* W_dec = (const float*)d_in[3];
    const float* b_dec = (const float*)d_in[4];
    const int*   pi_x  = (const int*)d_in[5];
    const int*   pi_z  = (const int*)d_in[6];

    float* preds = (float*)d_out;                     // [16384*1024]
    float* loss  = preds + (size_t)NPTS * DDIM;       // scalar tail
    float* Z     = (float*)d_ws;                      // 4 MB scratch

    // Encoder: 16384 rows / 128 per block
    encoder_kernel<<<NPTS / 128, 256, 0, stream>>>(X, W_enc, b_enc, Z);

    // Decoder: (16384/128 row blocks) x (1024/64 col groups) = 2048 blocks
    decoder_kernel<<<(NPTS / 128) * (DDIM / 64), 256, 0, stream>>>(Z, W_dec, b_dec, preds);

    // Loss: zero accumulator, then 65536 edges, 8 waves per block
    zero_loss_kernel<<<1, 1, 0, stream>>>(loss);
    topo_loss_kernel<<<(2 * EEDGES) / 8, 256, 0, stream>>>(X, Z, pi_x, pi_z, loss);
}